// GATRegression_40441412059604
// MI455X (gfx1250) — compile-verified
//
#include <hip/hip_runtime.h>
#include <hip/hip_bf16.h>
#include <math.h>

// ---------------------------------------------------------------------------
// GATRegression on MI455X (gfx1250, wave32).
// Dense GEMMs use V_WMMA_F32_16X16X4_F32 (exact fp32 matrix op) since the
// workload is scatter/gather bound, not GEMM bound. Edge softmax/aggregation
// use L2-resident atomics (feature tables fit in the 192MB L2).
// A-tiles are staged into LDS with GLOBAL_LOAD_ASYNC_TO_LDS_B128 (ASYNCcnt).
// ---------------------------------------------------------------------------

typedef __attribute__((ext_vector_type(2))) float v2f;
typedef __attribute__((ext_vector_type(8))) float v8f;
typedef __attribute__((ext_vector_type(4))) int   v4i;

#define IN_C   128
#define HID    64
#define HEADS  4
#define F1     (HEADS * HID)   // 256
#define NEG_SLOPE 0.2f
#define KC     32              // K-chunk staged in LDS for the A tile

// Async global->LDS staging (CDNA5 ASYNCcnt path), guarded so either
// toolchain compiles: fall back to synchronous staging if builtins absent.
#if defined(__has_builtin)
#  if __has_builtin(__builtin_amdgcn_global_load_async_to_lds_b128) && \
      __has_builtin(__builtin_amdgcn_s_wait_asynccnt)
#    define USE_ASYNC_LDS 1
#  endif
#endif
#ifndef USE_ASYNC_LDS
#  define USE_ASYNC_LDS 0
#endif

#if USE_ASYNC_LDS
typedef v4i __attribute__((address_space(1)))* gptr_v4i;   // global (AS1)
typedef v4i __attribute__((address_space(3)))* lptr_v4i;   // LDS (AS3)
#endif

static inline int cdiv(long long a, long long b) { return (int)((a + b - 1) / b); }

// ---------------- generic fill / broadcast / elu (b128 vectorized) --------
__global__ void fill_f32(float* __restrict__ p, float v, int n) {
    int i = blockIdx.x * blockDim.x + threadIdx.x;
    if (i < n) p[i] = v;
}

__global__ void fill_f32v4(float4* __restrict__ p, float v, int n4) {
    int i = blockIdx.x * blockDim.x + threadIdx.x;
    if (i < n4) p[i] = make_float4(v, v, v, v);
}

// out[n][c] = row[c]; cols4mask = (cols/4)-1, power-of-two columns
__global__ void bcast_rows4(float4* __restrict__ out, const float4* __restrict__ row,
                            int n4, int cols4mask) {
    int i = blockIdx.x * blockDim.x + threadIdx.x;
    if (i < n4) out[i] = row[i & cols4mask];
}

__device__ __forceinline__ float elu1(float v) { return (v > 0.f) ? v : expm1f(v); }

__global__ void elu4_kernel(float4* __restrict__ p, int n4) {
    int i = blockIdx.x * blockDim.x + threadIdx.x;
    if (i < n4) {
        float4 v = p[i];
        v.x = elu1(v.x); v.y = elu1(v.y); v.z = elu1(v.z); v.w = elu1(v.w);
        p[i] = v;
    }
}

// ---------------- fp32 WMMA GEMM: C[M,N] = A[M,K] @ B[K,N] ----------------
// One wave -> one 16x16 C tile. blockDim.x = 128 (4 waves); the 4 waves share
// the same 16 rows of A (staged through LDS) and cover 4 adjacent N-tiles.
__global__ void wmma_gemm_f32(const float* __restrict__ A, const float* __restrict__ B,
                              float* __restrict__ C, int M, int N, int K) {
    __shared__ float As[16 * KC];

    const int lane = threadIdx.x & 31;
    const int wave = threadIdx.x >> 5;
    const int tm   = blockIdx.x;                 // M-tile index
    const int tn   = blockIdx.y * 4 + wave;      // N-tile index
    const bool nOK = (tn * 16) < N;              // wave-uniform

    const int mrow = lane & 15;                  // A row within tile
    const int koff = (lane < 16) ? 0 : 2;        // f32 A/B: lanes 16-31 hold K+2,K+3
    const int ncol = tn * 16 + (lane & 15);

    v8f c = {};
    for (int k0 = 0; k0 < K; k0 += KC) {
        __syncthreads();                          // previous chunk fully consumed
#if USE_ASYNC_LDS
        {   // 128 threads x 16B = the whole 16xKC tile in one async b128 wavefront
            int r   = threadIdx.x >> 3;           // 0..15 (tile row)
            int seg = threadIdx.x & 7;            // 0..7  (KC floats = 8 x float4)
            int gr  = tm * 16 + r; gr = gr < M ? gr : (M - 1);
            const float* gsrc = A + (size_t)gr * K + k0 + seg * 4;
            __builtin_amdgcn_global_load_async_to_lds_b128(
                (gptr_v4i)gsrc, (lptr_v4i)&As[r * KC + seg * 4], 0, 0);
        }
        __builtin_amdgcn_s_wait_asynccnt(0);
        __syncthreads();
#else
        for (int idx = threadIdx.x; idx < 16 * KC; idx += blockDim.x) {
            int r  = idx / KC;
            int kk = idx % KC;
            int gr = tm * 16 + r;
            gr = gr < M ? gr : (M - 1);
            As[idx] = A[(size_t)gr * K + k0 + kk];
        }
        __syncthreads();
#endif
        if (nOK) {
#pragma unroll
            for (int ks = 0; ks < KC; ks += 4) {
                v2f a, b;
                a.x = As[mrow * KC + ks + koff];
                a.y = As[mrow * KC + ks + koff + 1];
                b.x = B[(size_t)(k0 + ks + koff) * N + ncol];
                b.y = B[(size_t)(k0 + ks + koff + 1) * N + ncol];
                // D = A*B + C  (fp32 matrix pipe)
                c = __builtin_amdgcn_wmma_f32_16x16x4_f32(
                        false, a, false, b, (short)0, c, false, false);
            }
        }
    }
    if (!nOK) return;
    // C/D layout: VGPR r -> M = r (lanes 0-15) or r+8 (lanes 16-31), N = lane&15
    const int rbase = (lane < 16) ? 0 : 8;
    float* Cp = C + (size_t)(tm * 16) * N + ncol; // per-lane base; offsets are imm
    if (tm * 16 + 15 < M) {                       // wave-uniform full-tile fast path
#pragma unroll
        for (int r = 0; r < 8; ++r)
            Cp[(rbase + r) * N] = c[r];
    } else {
#pragma unroll
        for (int r = 0; r < 8; ++r)
            if (tm * 16 + rbase + r < M) Cp[(rbase + r) * N] = c[r];
    }
}

// ---------------- per-(node, head) attention dot products ----------------
template <int H>
__global__ void alpha_kernel_t(const float* __restrict__ h,
                               const float* __restrict__ att_s,
                               const float* __restrict__ att_d,
                               float* __restrict__ as, float* __restrict__ ad, int N) {
    int id = blockIdx.x * blockDim.x + threadIdx.x;
    if (id >= N * H) return;
    int n  = id / H;           // H is a power-of-two constant -> shifts
    int hd = id % H;
    const float* hp = h + (size_t)n * H * HID + (size_t)hd * HID;
    const float* sp = att_s + hd * HID;
    const float* dp = att_d + hd * HID;
    float s1 = 0.f, s2 = 0.f;
#pragma unroll 8
    for (int c = 0; c < HID; ++c) {
        float v = hp[c];
        s1 += v * sp[c];
        s2 += v * dp[c];
    }
    as[id] = s1;
    ad[id] = s2;
}

// ---------------- edge pass A: exp(leaky_relu(e)) + atomic denom ----------
// Softmax is shift invariant and logits are O(1) here, so the segment-max
// pass is elided. Numerators are cached in `ex` for pass B.
template <int H>
__global__ void edge_softmax_num_t(const int* __restrict__ ei, int E, int N,
                                   const float* __restrict__ asrc,
                                   const float* __restrict__ adst,
                                   float* __restrict__ ex, float* __restrict__ denom) {
    int id = blockIdx.x * blockDim.x + threadIdx.x;  // (E+N)*H < 2^31
    if (id >= (E + N) * H) return;
    int e  = id / H;
    int hd = id % H;
    int s  = (e < E) ? ei[e]     : (e - E);   // self loops appended after edges
    int d  = (e < E) ? ei[E + e] : (e - E);
    float v = asrc[(size_t)s * H + hd] + adst[(size_t)d * H + hd];
    v = (v > 0.f) ? v : NEG_SLOPE * v;
    float xv = expf(v);
    ex[(size_t)e * H + hd] = xv;
    atomicAdd(&denom[(size_t)d * H + hd], xv);
}

// ---------------- edge pass B: out[d] += (ex/denom) * h[s]  ----------------
// One wave per edge; per-head weights hoisted to registers; 32-lane coalesced
// strips over F = H*64 features.
template <int H>
__global__ void edge_aggregate_t(const int* __restrict__ ei, int E, int N,
                                 const float* __restrict__ ex,
                                 const float* __restrict__ denom,
                                 const float* __restrict__ h,
                                 float* __restrict__ out) {
    int wid  = (int)(((long long)blockIdx.x * blockDim.x + threadIdx.x) >> 5);
    int lane = threadIdx.x & 31;
    if (wid >= E + N) return;
    int s = (wid < E) ? ei[wid]     : (wid - E);
    int d = (wid < E) ? ei[E + wid] : (wid - E);
    float wq[H];
#pragma unroll
    for (int hd = 0; hd < H; ++hd)
        wq[hd] = ex[(size_t)wid * H + hd] / (denom[(size_t)d * H + hd] + 1e-16f);
    const float* hs = h   + (size_t)s * (H * HID);
    float*       od = out + (size_t)d * (H * HID);
#pragma unroll
    for (int i = 0; i < (H * HID) / 32; ++i) {
        int c = lane + 32 * i;
        atomicAdd(&od[c], wq[c >> 6] * hs[c]);   // HID == 64
    }
}

// ---------------- fused ELU + global mean-pool (numerators) ----------------
__global__ void pool_kernel(const float* __restrict__ h, const int* __restrict__ batch,
                            float* __restrict__ sums, float* __restrict__ cnt, int N) {
    int i = blockIdx.x * blockDim.x + threadIdx.x;
    if (i >= N * HID) return;
    int n = i >> 6;          // HID == 64
    int c = i & 63;
    float v = elu1(h[i]);
    int b = batch[n];
    atomicAdd(&sums[(size_t)b * HID + c], v);
    if (c == 0) atomicAdd(&cnt[b], 1.0f);
}

// ---------------- MLP head: one block per graph ----------------
__global__ void mlp_kernel(const float* __restrict__ sums, const float* __restrict__ cnt,
                           const float* __restrict__ w1, const float* __restrict__ b1,
                           const float* __restrict__ w2, const float* __restrict__ b2,
                           float* __restrict__ out) {
    __shared__ float pooled[HID];
    __shared__ float red[128];
    int g = blockIdx.x;
    int t = threadIdx.x;   // 128 threads = hidden width of lin1
    float c = fmaxf(cnt[g], 1.0f);
    if (t < HID) pooled[t] = sums[(size_t)g * HID + t] / c;
    __syncthreads();
    float acc = b1[t];
#pragma unroll 8
    for (int i = 0; i < HID; ++i) acc += pooled[i] * w1[i * 128 + t];
    float z = fmaxf(acc, 0.f);          // relu
    red[t] = z * w2[t];                 // lin2_w is [128,1]
    __syncthreads();
    for (int s = 64; s > 0; s >>= 1) {
        if (t < s) red[t] += red[t + s];
        __syncthreads();
    }
    if (t == 0) out[g] = red[0] + b2[0];
}

// ---------------------------------------------------------------------------
extern "C" void kernel_launch(void* const* d_in, const int* in_sizes, int n_in,
                              void* d_out, int out_size, void* d_ws, size_t ws_size,
                              hipStream_t stream) {
    const float* x        = (const float*)d_in[0];
    const int*   ei       = (const int*)d_in[1];    // [2,E]
    const int*   batch    = (const int*)d_in[2];    // [N]
    const float* W1       = (const float*)d_in[3];  // [128,256]
    const float* att_src1 = (const float*)d_in[4];  // [4,64]
    const float* att_dst1 = (const float*)d_in[5];
    const float* b1       = (const float*)d_in[6];  // [256]
    const float* W2       = (const float*)d_in[7];  // [256,64]
    const float* att_src2 = (const float*)d_in[8];  // [1,64]
    const float* att_dst2 = (const float*)d_in[9];
    const float* b2       = (const float*)d_in[10]; // [64]
    const float* lin1_w   = (const float*)d_in[11]; // [64,128]
    const float* lin1_b   = (const float*)d_in[12]; // [128]
    const float* lin2_w   = (const float*)d_in[13]; // [128,1]
    const float* lin2_b   = (const float*)d_in[14]; // [1]
    float* out = (float*)d_out;

    const int N = in_sizes[0] / IN_C;
    const int E = in_sizes[1] / 2;
    const int G = out_size;
    const int EN = E + N;

    // ----- workspace carve-up (fp32) -----
    float* w = (float*)d_ws;
    size_t off = 0;
    float* h1   = w + off; off += (size_t)N * F1;     // pre-agg features L1
    float* out1 = w + off; off += (size_t)N * F1;     // aggregated L1 (init=b1)
    float* as1  = w + off; off += (size_t)N * HEADS;
    float* ad1  = w + off; off += (size_t)N * HEADS;
    float* dn1  = w + off; off += (size_t)N * HEADS;
    float* ex1  = w + off; off += (size_t)EN * HEADS;
    float* h2   = w + off; off += (size_t)N * HID;
    float* out2 = w + off; off += (size_t)N * HID;    // aggregated L2 (init=b2)
    float* as2  = w + off; off += (size_t)N;
    float* ad2  = w + off; off += (size_t)N;
    float* dn2  = w + off; off += (size_t)N;
    float* ex2  = w + off; off += (size_t)EN;
    float* sums = w + off; off += (size_t)G * HID;
    float* cnt  = w + off; off += (size_t)G;
    (void)ws_size;

    const int T = 256;

    // ----- init accumulators (ws is poisoned; must init every call) -----
    fill_f32v4<<<cdiv(N * HEADS / 4, T), T, 0, stream>>>((float4*)dn1, 0.f, N * HEADS / 4);
    fill_f32<<<cdiv(N, T), T, 0, stream>>>(dn2, 0.f, N);
    fill_f32v4<<<cdiv(G * HID / 4, T), T, 0, stream>>>((float4*)sums, 0.f, G * HID / 4);
    fill_f32<<<cdiv(G, T), T, 0, stream>>>(cnt, 0.f, G);
    bcast_rows4<<<cdiv((long long)N * F1 / 4, T), T, 0, stream>>>(
        (float4*)out1, (const float4*)b1, N * (F1 / 4), (F1 / 4) - 1);
    bcast_rows4<<<cdiv((long long)N * HID / 4, T), T, 0, stream>>>(
        (float4*)out2, (const float4*)b2, N * (HID / 4), (HID / 4) - 1);

    // ----- layer 1 -----
    {   // h1 = x @ W1        (M=N, K=128, Ncols=256)
        dim3 grid(cdiv(N, 16), F1 / 64);   // 4 N-tiles per block
        wmma_gemm_f32<<<grid, 128, 0, stream>>>(x, W1, h1, N, F1, IN_C);
    }
    alpha_kernel_t<HEADS><<<cdiv((long long)N * HEADS, T), T, 0, stream>>>(
        h1, att_src1, att_dst1, as1, ad1, N);
    edge_softmax_num_t<HEADS><<<cdiv((long long)EN * HEADS, T), T, 0, stream>>>(
        ei, E, N, as1, ad1, ex1, dn1);
    edge_aggregate_t<HEADS><<<cdiv(EN, T / 32), T, 0, stream>>>(
        ei, E, N, ex1, dn1, h1, out1);
    elu4_kernel<<<cdiv((long long)N * F1 / 4, T), T, 0, stream>>>(
        (float4*)out1, N * (F1 / 4));

    // ----- layer 2 -----
    {   // h2 = elu(out1) @ W2   (M=N, K=256, Ncols=64)
        dim3 grid(cdiv(N, 16), 1);
        wmma_gemm_f32<<<grid, 128, 0, stream>>>(out1, W2, h2, N, HID, F1);
    }
    alpha_kernel_t<1><<<cdiv(N, T), T, 0, stream>>>(h2, att_src2, att_dst2, as2, ad2, N);
    edge_softmax_num_t<1><<<cdiv(EN, T), T, 0, stream>>>(ei, E, N, as2, ad2, ex2, dn2);
    edge_aggregate_t<1><<<cdiv(EN, T / 32), T, 0, stream>>>(ei, E, N, ex2, dn2, h2, out2);

    // ----- fused ELU + mean pool, then MLP head -----
    pool_kernel<<<cdiv((long long)N * HID, T), T, 0, stream>>>(out2, batch, sums, cnt, N);
    mlp_kernel<<<G, 128, 0, stream>>>(sums, cnt, lin1_w, lin1_b, lin2_w, lin2_b, out);
}